// SpatialTransformBlock_65240553226712
// MI455X (gfx1250) — compile-verified
//
#include <hip/hip_runtime.h>

// BEV grid-sample (spatial transformer) for MI455X / gfx1250, wave32.
// - Projective transform via V_WMMA_F32_16X16X4_F32 (f32 WMMA: coordinates
//   need full f32 precision; D = theta(16x4) x base(4x16) puts each pixel's
//   (g0,g1,g2) directly into that lane's d[0..2]).
// - Bandwidth-bound: coalesced per-channel nontemporal stores, clustered
//   bilinear gathers stay in the 192MB L2.

typedef __attribute__((ext_vector_type(2))) float v2f;
typedef __attribute__((ext_vector_type(8))) float v8f;

namespace {
constexpr int kN    = 4;
constexpr int kC    = 16;
constexpr int kHin  = 288;
constexpr int kWin  = 1216;
constexpr int kHout = 800;
constexpr int kWout = 400;
constexpr int kHWout = kHout * kWout;   // 320000  (divisible by 256)
constexpr int kHWin  = kHin * kWin;     // 350208
constexpr int kBlock = 256;             // 8 waves of 32
constexpr int kPixels = kN * kHWout;    // 1,280,000 = 5000 * 256 exactly
}

__global__ __launch_bounds__(kBlock)
void bev_grid_sample_kernel(const float* __restrict__ x,
                            const float* __restrict__ theta,
                            const float* __restrict__ shift,
                            float* __restrict__ out) {
  const int tid  = blockIdx.x * kBlock + threadIdx.x;   // == global pixel id
  const int lane = threadIdx.x & 31;                    // wave32
  const int m    = lane & 15;
  const bool hi  = lane >= 16;

  const int p   = tid;
  const int n   = p / kHWout;               // uniform per block (kHWout%256==0)
  const int rem = p - n * kHWout;
  const int h   = rem / kWout;
  const int w   = rem - h * kWout;

  const float gr = 0.05f * (800.0f / (float)kHout);
  const float xoff = -10.0f + 0.5f * gr;
  const float yoff = 46.0f - 0.5f * gr;

  // ---------------- projective transform: g = theta @ [bx, by, 1] ----------
  float g0, g1, g2;

#if __has_builtin(__builtin_amdgcn_wmma_f32_16x16x4_f32)
  // A = theta rows (16x4, rows>=3 and K=3 padded with 0).
  // ISA f32 A layout: lanes 0-15 hold K=0 (v0), K=1 (v1); lanes 16-31 hold K=2,K=3.
  const float* th = theta + n * 9;
  v2f A;
  A.x = (m < 3) ? th[m * 3 + (hi ? 2 : 0)] : 0.0f;
  A.y = (m < 3 && !hi) ? th[m * 3 + 1] : 0.0f;

  // B = base coords (4x16 pixels). Column n' = pixel waveBase + q*16 + n'.
  // Lanes 0-15: B[0][n']=bx (v0), B[1][n']=by (v1); lanes 16-31: B[2][n']=1, B[3][n']=0.
  const int waveBase = p - lane;
  v2f B0, B1;
  {
    int pq = waveBase + m;                  // pixels 0..15 of this wave
    int rq = pq - n * kHWout;
    int hq = rq / kWout;
    int wq = rq - hq * kWout;
    B0.x = hi ? 1.0f : (xoff + gr * (float)wq);
    B0.y = hi ? 0.0f : (yoff - gr * (float)hq);

    pq = waveBase + 16 + m;                 // pixels 16..31 of this wave
    rq = pq - n * kHWout;
    hq = rq / kWout;
    wq = rq - hq * kWout;
    B1.x = hi ? 1.0f : (xoff + gr * (float)wq);
    B1.y = hi ? 0.0f : (yoff - gr * (float)hq);
  }

  v8f Z = {};
  // 8 args: (neg_a, A, neg_b, B, c_mod, C, reuse_a, reuse_b)
  v8f D0 = __builtin_amdgcn_wmma_f32_16x16x4_f32(false, A, false, B0,
                                                 (short)0, Z, false, false);
  v8f D1 = __builtin_amdgcn_wmma_f32_16x16x4_f32(false, A, false, B1,
                                                 (short)0, Z, false, false);

  // D[r][n'] lives in lane n' (0-15), element r. Lane L wants pixel waveBase+L:
  // lower half from D0 lane L, upper half from D1 lane L-16. Full-EXEC shuffles.
  const float s00 = __shfl(D0[0], m), s01 = __shfl(D0[1], m), s02 = __shfl(D0[2], m);
  const float s10 = __shfl(D1[0], m), s11 = __shfl(D1[1], m), s12 = __shfl(D1[2], m);
  g0 = hi ? s10 : s00;
  g1 = hi ? s11 : s01;
  g2 = hi ? s12 : s02;
#else
#warning "wmma_f32_16x16x4_f32 builtin unavailable; scalar fallback in use"
  const float* th = theta + n * 9;
  const float bx = xoff + gr * (float)w;
  const float by = yoff - gr * (float)h;
  g0 = th[0] * bx + th[1] * by + th[2];
  g1 = th[3] * bx + th[4] * by + th[5];
  g2 = th[6] * bx + th[7] * by + th[8];
#endif

  // ---------------- normalized grid -> input pixel coords ------------------
  const float invz = 1.0f / g2;
  const float px = g0 * invz;
  const float py = g1 * invz;
  const float gx = px * (1.0f / 608.0f) - 1.0f;
  const float gy = (py - shift[n]) * (1.0f / 144.0f) - 1.0f;
  const float ix = ((gx + 1.0f) * (float)kWin - 1.0f) * 0.5f;
  const float iy = ((gy + 1.0f) * (float)kHin - 1.0f) * 0.5f;

  const float fx0 = floorf(ix);
  const float fy0 = floorf(iy);
  const float wx = ix - fx0;
  const float wy = iy - fy0;
  const int x0 = (int)fminf(fmaxf(fx0,        0.0f), (float)(kWin - 1));
  const int x1 = (int)fminf(fmaxf(fx0 + 1.0f, 0.0f), (float)(kWin - 1));
  const int y0 = (int)fminf(fmaxf(fy0,        0.0f), (float)(kHin - 1));
  const int y1 = (int)fminf(fmaxf(fy0 + 1.0f, 0.0f), (float)(kHin - 1));

  const float w00 = (1.0f - wx) * (1.0f - wy);
  const float w01 = wx * (1.0f - wy);
  const float w10 = (1.0f - wx) * wy;
  const float w11 = wx * wy;

  const int i00 = y0 * kWin + x0;
  const int i01 = y0 * kWin + x1;
  const int i10 = y1 * kWin + x0;
  const int i11 = y1 * kWin + x1;

  const float* __restrict__ xb = x + (size_t)n * kC * kHWin;
  float* __restrict__ ob = out + (size_t)n * kC * kHWout + (size_t)h * kWout + w;

  // 16 channels share the same 4 gather indices and weights; stores are
  // coalesced across lanes (consecutive w) and bypass L2 (write-once stream).
#pragma unroll
  for (int c = 0; c < kC; ++c) {
    const float* __restrict__ xc = xb + (size_t)c * kHWin;
    const float v = xc[i00] * w00 + xc[i01] * w01 + xc[i10] * w10 + xc[i11] * w11;
    __builtin_nontemporal_store(v, ob + (size_t)c * kHWout);
  }
}

extern "C" void kernel_launch(void* const* d_in, const int* in_sizes, int n_in,
                              void* d_out, int out_size, void* d_ws, size_t ws_size,
                              hipStream_t stream) {
  (void)in_sizes; (void)n_in; (void)out_size; (void)d_ws; (void)ws_size;
  const float* x     = (const float*)d_in[0];  // (4,16,288,1216) f32
  const float* theta = (const float*)d_in[1];  // (4,3,3) f32
  const float* shift = (const float*)d_in[2];  // (4,) f32
  // d_in[3], d_in[4] are H_out/W_out scalars (800, 400) — baked in as constants.
  float* out = (float*)d_out;                  // (4,16,800,400) f32

  const int blocks = kPixels / kBlock;         // 5000, exact cover (no tail guard,
                                               // EXEC all-ones for WMMA)
  bev_grid_sample_kernel<<<blocks, kBlock, 0, stream>>>(x, theta, shift, out);
}